// BiPartialTestBlock_59854664237110
// MI455X (gfx1250) — compile-verified
//
#include <hip/hip_runtime.h>
#include <hip/hip_bf16.h>
#include <math.h>

// ---------------------------------------------------------------------------
// Problem constants (fixed by the reference setup_inputs()).
// ---------------------------------------------------------------------------
constexpr int BQ  = 2;
constexpr int NLQ = 16384;
constexpr int NCQ = 8192;
constexpr int DQ  = 64;
constexpr int HQ  = 128;

typedef __attribute__((ext_vector_type(2))) float v2f;
typedef __attribute__((ext_vector_type(8))) float v8f;

// ---------------------------------------------------------------------------
// WMMA GEMM over the big Ms tensor (fp32, f32-accumulate, 16x16x4 tiles).
//
//   TRANSA = true :  Out[c][d] = sum_l Ms[l][c] * Bm[l][d]   (K = nL, Mtot = nC)
//   TRANSA = false:  Out[l][d] = sum_c Ms[l][c] * Bm[c][d]   (K = nC, Mtot = nL)
//
// Block = 128 threads = 4 waves. Each wave: 16 M-rows x 64 N-cols
// (4 accumulator tiles). Lane mapping per ISA layouts:
//   A (16x4 f32): lane m = lane&15, regs hold K = kb, kb+1 with kb = (lane>>4)*2
//   B (4x16 f32): lane n = lane&15, regs hold K = kb, kb+1
//   C/D (16x16) : VGPR g -> row m0 + g + (lane>>4)*8, col = lane&15
//
// Ms (1.07 GB, streamed once per pass) is loaded NON-TEMPORAL so it does not
// evict the L2-resident B operand (4 MB/batch) and MLP weights. The prefetch
// runs a few hundred K ahead of the NT consume.
// ---------------------------------------------------------------------------
template <bool TRANSA>
__global__ __launch_bounds__(128) void gemm_ms_wmma(
    const float* __restrict__ Ms, const float* __restrict__ Bm,
    float* __restrict__ Out, int K, int Mtot) {
  const int lane = threadIdx.x & 31;
  const int wave = threadIdx.x >> 5;
  const int n    = lane & 15;
  const int kb   = (lane >> 4) << 1;  // 0 or 2
  const int m0   = blockIdx.x * 64 + wave * 16;

  const size_t batch = blockIdx.y;
  const float* __restrict__ Ab = Ms + batch * (size_t)NLQ * (size_t)NCQ;
  const float* __restrict__ Bb = Bm + batch * (size_t)K * DQ;
  float* __restrict__ Ob       = Out + batch * (size_t)Mtot * DQ;

  v8f acc[4];
  acc[0] = v8f{}; acc[1] = v8f{}; acc[2] = v8f{}; acc[3] = v8f{};

  for (int k = 0; k < K; k += 4) {
    v2f a;
    if (TRANSA) {
      // A[m][kk] = Ms[(k+kk)*NC + m0 + m]; lanes 0-15 read 64B contiguous.
      const size_t r = (size_t)(k + kb) * NCQ + (size_t)(m0 + n);
      a.x = __builtin_nontemporal_load(Ab + r);
      a.y = __builtin_nontemporal_load(Ab + r + NCQ);
      const int kpf = (k + 256 < K) ? (k + 256) : k;
      __builtin_prefetch(Ab + (size_t)(kpf + kb) * NCQ + (size_t)(m0 + n), 0, 3);
    } else {
      // A[m][kk] = Ms[(m0+m)*NC + k + kk]; consecutive k -> adjacent loads.
      const float* __restrict__ ap = Ab + (size_t)(m0 + n) * NCQ + (size_t)(k + kb);
      a.x = __builtin_nontemporal_load(ap);
      a.y = __builtin_nontemporal_load(ap + 1);
      const int kpf = (k + 512 < K) ? (k + 512) : k;
      __builtin_prefetch(Ab + (size_t)(m0 + n) * NCQ + (size_t)(kpf + kb), 0, 3);
    }

    const size_t b0 = (size_t)(k + kb) * DQ + (size_t)n;  // row k+kb, col n
#pragma unroll
    for (int j = 0; j < 4; ++j) {
      v2f bb;
      bb.x = Bb[b0 + j * 16];       // K = k+kb
      bb.y = Bb[b0 + j * 16 + DQ];  // K = k+kb+1 (next row, stride 64 floats)
      acc[j] = __builtin_amdgcn_wmma_f32_16x16x4_f32(
          false, a, false, bb, (short)0, acc[j], false, false);
    }
  }

  const int rbase = m0 + ((lane >> 4) << 3);
#pragma unroll
  for (int g = 0; g < 8; ++g) {
    const size_t ro = (size_t)(rbase + g) * DQ + (size_t)n;
    Ob[ro + 0]  = acc[0][g];
    Ob[ro + 16] = acc[1][g];
    Ob[ro + 32] = acc[2][g];
    Ob[ro + 48] = acc[3][g];
  }
}

// ---------------------------------------------------------------------------
// Per-token MLP helpers. Weight pointers are lane-uniform -> scalar (K$) loads.
// Layer 1 iterates input dim outer so W1 rows (contiguous 512B) stream nicely.
// ---------------------------------------------------------------------------
__device__ __forceinline__ void l1_accum(float h[HQ], const float* __restrict__ src,
                                         const float* __restrict__ w1seg, int cnt) {
  for (int i = 0; i < cnt; ++i) {
    const float xi = src[i];
    const float* __restrict__ w = w1seg + i * HQ;
#pragma unroll
    for (int j = 0; j < HQ; ++j) h[j] = __builtin_fmaf(xi, w[j], h[j]);
  }
}

__device__ __forceinline__ void l2_out(const float h[HQ],
                                       const float* __restrict__ W2,
                                       const float* __restrict__ B2, float y[DQ]) {
#pragma unroll
  for (int k = 0; k < DQ; ++k) y[k] = B2[k];
  for (int j = 0; j < HQ; ++j) {
    const float hj = fmaxf(h[j], 0.0f);
    const float* __restrict__ w = W2 + j * DQ;
#pragma unroll
    for (int k = 0; k < DQ; ++k) y[k] = __builtin_fmaf(hj, w[k], y[k]);
  }
}

// Y = MLP64(X)   (used for Lmsg = MLP(Ls) and Cmsg = MLP(cu))
__global__ __launch_bounds__(256) void mlp64_kernel(
    const float* __restrict__ X, const float* __restrict__ W1,
    const float* __restrict__ B1, const float* __restrict__ W2,
    const float* __restrict__ B2, float* __restrict__ Y, int ntok) {
  const int tok = blockIdx.x * blockDim.x + threadIdx.x;
  if (tok >= ntok) return;
  float h[HQ];
#pragma unroll
  for (int j = 0; j < HQ; ++j) h[j] = B1[j];
  l1_accum(h, X + (size_t)tok * DQ, W1, DQ);
  float y[DQ];
  l2_out(h, W2, B2, y);
  float* __restrict__ yp = Y + (size_t)tok * DQ;
#pragma unroll
  for (int k = 0; k < DQ; ++k) yp[k] = y[k];
}

// cu = Cs + tanh(MLP(concat[Cs, MTLmsg]))   (IN = 128)
__global__ __launch_bounds__(256) void cu_update_kernel(
    const float* __restrict__ Cs, const float* __restrict__ MT,
    const float* __restrict__ W1, const float* __restrict__ B1,
    const float* __restrict__ W2, const float* __restrict__ B2,
    float* __restrict__ cu, int ntok) {
  const int tok = blockIdx.x * blockDim.x + threadIdx.x;
  if (tok >= ntok) return;
  float h[HQ];
#pragma unroll
  for (int j = 0; j < HQ; ++j) h[j] = B1[j];
  l1_accum(h, Cs + (size_t)tok * DQ, W1, DQ);
  l1_accum(h, MT + (size_t)tok * DQ, W1 + DQ * HQ, DQ);
  float y[DQ];
  l2_out(h, W2, B2, y);
  const float* __restrict__ cs = Cs + (size_t)tok * DQ;
  float* __restrict__ o        = cu + (size_t)tok * DQ;
#pragma unroll
  for (int k = 0; k < DQ; ++k) o[k] = cs[k] + tanhf(y[k]);
}

// lu = Ls + tanh(MLP(concat[Ls, flip(Ls), MCmsg]))   (IN = 192, flip = tok^1)
__global__ __launch_bounds__(256) void lu_update_kernel(
    const float* __restrict__ Ls, const float* __restrict__ MC,
    const float* __restrict__ W1, const float* __restrict__ B1,
    const float* __restrict__ W2, const float* __restrict__ B2,
    float* __restrict__ lu, int ntok) {
  const int tok = blockIdx.x * blockDim.x + threadIdx.x;
  if (tok >= ntok) return;
  const int ptok = tok ^ 1;  // pairs stay inside a batch (nL even)
  float h[HQ];
#pragma unroll
  for (int j = 0; j < HQ; ++j) h[j] = B1[j];
  l1_accum(h, Ls + (size_t)tok * DQ,  W1,               DQ);
  l1_accum(h, Ls + (size_t)ptok * DQ, W1 + DQ * HQ,     DQ);
  l1_accum(h, MC + (size_t)tok * DQ,  W1 + 2 * DQ * HQ, DQ);
  float y[DQ];
  l2_out(h, W2, B2, y);
  const float* __restrict__ ls = Ls + (size_t)tok * DQ;
  float* __restrict__ o        = lu + (size_t)tok * DQ;
#pragma unroll
  for (int k = 0; k < DQ; ++k) o[k] = ls[k] + tanhf(y[k]);
}

// ---------------------------------------------------------------------------
// Launcher
// ---------------------------------------------------------------------------
extern "C" void kernel_launch(void* const* d_in, const int* in_sizes, int n_in,
                              void* d_out, int out_size, void* d_ws, size_t ws_size,
                              hipStream_t stream) {
  (void)in_sizes; (void)n_in; (void)out_size; (void)ws_size;

  const float* Ls    = (const float*)d_in[0];
  const float* Cs    = (const float*)d_in[1];
  const float* Ms    = (const float*)d_in[2];
  const float* Lm_w1 = (const float*)d_in[3];
  const float* Lm_b1 = (const float*)d_in[4];
  const float* Lm_w2 = (const float*)d_in[5];
  const float* Lm_b2 = (const float*)d_in[6];
  const float* Cm_w1 = (const float*)d_in[7];
  const float* Cm_b1 = (const float*)d_in[8];
  const float* Cm_w2 = (const float*)d_in[9];
  const float* Cm_b2 = (const float*)d_in[10];
  const float* Cu_w1 = (const float*)d_in[11];
  const float* Cu_b1 = (const float*)d_in[12];
  const float* Cu_w2 = (const float*)d_in[13];
  const float* Cu_b2 = (const float*)d_in[14];
  const float* Lu_w1 = (const float*)d_in[15];
  const float* Lu_b1 = (const float*)d_in[16];
  const float* Lu_w2 = (const float*)d_in[17];
  const float* Lu_b2 = (const float*)d_in[18];

  const size_t nLtok = (size_t)BQ * NLQ;  // 32768
  const size_t nCtok = (size_t)BQ * NCQ;  // 16384

  float* lu = (float*)d_out;
  float* cu = (float*)d_out + nLtok * DQ;

  // Workspace (floats): [Lmsg | MTLmsg | Cmsg]; MCmsg aliases Lmsg (dead after
  // the first GEMM). Total = 4 Mi floats = 16 MB.
  float* ws    = (float*)d_ws;
  float* Lmsg  = ws;                              // B*NL*D
  float* MT    = ws + nLtok * DQ;                 // B*NC*D
  float* Cmsg  = ws + nLtok * DQ + nCtok * DQ;    // B*NC*D
  float* MC    = Lmsg;                            // alias: B*NL*D

  // 1) Lmsg = MLP(Ls)
  mlp64_kernel<<<dim3((unsigned)(nLtok / 256)), dim3(256), 0, stream>>>(
      Ls, Lm_w1, Lm_b1, Lm_w2, Lm_b2, Lmsg, (int)nLtok);

  // 2) MTLmsg = Ms^T @ Lmsg   (K = nL, Mtot = nC)
  gemm_ms_wmma<true><<<dim3(NCQ / 64, BQ), dim3(128), 0, stream>>>(
      Ms, Lmsg, MT, NLQ, NCQ);

  // 3) cu = Cs + tanh(MLP(concat[Cs, MTLmsg]))
  cu_update_kernel<<<dim3((unsigned)(nCtok / 256)), dim3(256), 0, stream>>>(
      Cs, MT, Cu_w1, Cu_b1, Cu_w2, Cu_b2, cu, (int)nCtok);

  // 4) Cmsg = MLP(cu)
  mlp64_kernel<<<dim3((unsigned)(nCtok / 256)), dim3(256), 0, stream>>>(
      cu, Cm_w1, Cm_b1, Cm_w2, Cm_b2, Cmsg, (int)nCtok);

  // 5) MCmsg = Ms @ Cmsg   (K = nC, Mtot = nL)
  gemm_ms_wmma<false><<<dim3(NLQ / 64, BQ), dim3(128), 0, stream>>>(
      Ms, Cmsg, MC, NCQ, NLQ);

  // 6) lu = Ls + tanh(MLP(concat[Ls, flip(Ls), MCmsg]))
  lu_update_kernel<<<dim3((unsigned)(nLtok / 256)), dim3(256), 0, stream>>>(
      Ls, MC, Lu_w1, Lu_b1, Lu_w2, Lu_b2, lu, (int)nLtok);
}